// PointCloudCompletionLoss_28114855920184
// MI455X (gfx1250) — compile-verified
//
#include <hip/hip_runtime.h>

typedef __attribute__((ext_vector_type(2))) float v2f;
typedef __attribute__((ext_vector_type(8))) float v8f;

#define WAVES 4          // waves per block, each owns one 16-row X tile
#define CH    1024       // Y points staged in LDS per chunk (16 KB as float4)
#define TPC   (CH / 16)  // 16-point Y tiles per chunk

// Computes sum over rows n of min_m ||x_n - y_m||^2 for each batch, scaled by
// (*param) * invDenom, atomically accumulated into out[slot].
// Column-mins of the full Chamfer come from a second launch with X/Y swapped.
//
// Tricks:
//  1) Fold linear terms into the K=4 WMMA:
//       A row n = (-2x0, -2x1, -2x2, 1)
//       B col m = ( y0,   y1,   y2,  ||y||^2 )
//       => acc[n,m] = -2 x.y + ||y||^2
//  2) ||x||^2 is constant over m, and min commutes with adding a constant, so
//     the inner loop tracks min_m acc only; ||x||^2 is added once at the end.
//     Inner loop = ds_load_b64 + v_wmma + v_min only.
__global__ __launch_bounds__(128) void chamfer_rowmin_kernel(
    const float* __restrict__ X, const float* __restrict__ Y,
    int N, int M, const int* __restrict__ param, float invDenom,
    float* __restrict__ out, int slot)
{
    __shared__ float4 ylds[CH];          // {y0, y1, y2, |y|^2} per point
    __shared__ float  x2s[WAVES][16];

    const int tid  = threadIdx.x;
    const int wave = tid >> 5;
    const int lane = tid & 31;
    const int hi   = lane >> 4;   // lane half: 0 -> K{0,1}, 1 -> K{2,3}
    const int lr   = lane & 15;

    const int blocksPerBatch = N / (16 * WAVES);
    const int b       = blockIdx.x / blocksPerBatch;
    const int tile    = (blockIdx.x % blocksPerBatch) * WAVES + wave;
    const int rowbase = tile * 16;

    // ---- A matrix: 16x4 f32, lanes 0-15 hold {K0,K1}, lanes 16-31 {K2,K3} ----
    const float* xp = X + ((size_t)b * N + rowbase + lr) * 3;
    const float xc0 = xp[0], xc1 = xp[1], xc2 = xp[2];
    v2f a;
    a.x = hi ? (-2.0f * xc2) : (-2.0f * xc0);
    a.y = hi ? 1.0f          : (-2.0f * xc1);
    x2s[wave][lr] = xc0 * xc0 + xc1 * xc1 + xc2 * xc2;
    __syncthreads();

    float rmin[8];
#pragma unroll
    for (int v = 0; v < 8; ++v) rmin[v] = 3.0e38f;

    const float* Yb = Y + (size_t)b * M * 3;

    // per-lane B fetch base: point*16 bytes + half*8 (one ds_load_b64 per tile)
    const float* ybase = (const float*)ylds + hi * 2 + lr * 4;

    for (int m0 = 0; m0 < M; m0 += CH) {
        // cooperative stage: load y, precompute |y|^2, pack as float4 in LDS
        for (int p = tid; p < CH; p += 128) {
            const float* yy = Yb + (size_t)(m0 + p) * 3;
            const float c0 = yy[0], c1 = yy[1], c2 = yy[2];
            float4 q;
            q.x = c0; q.y = c1; q.z = c2;
            q.w = c0 * c0 + c1 * c1 + c2 * c2;
            ylds[p] = q;
        }
        if (m0 + CH < M)  // prefetch next chunk (global_prefetch_b8)
            __builtin_prefetch((const char*)(Yb + (size_t)(m0 + CH) * 3) + tid * 128, 0, 1);
        __syncthreads();

#pragma unroll 4
        for (int t = 0; t < TPC; ++t) {
            const v2f bv = *(const v2f*)(ybase + t * 64);   // ds_load_b64

            v8f acc = {};
            acc = __builtin_amdgcn_wmma_f32_16x16x4_f32(
                /*neg_a=*/false, a, /*neg_b=*/false, bv,
                /*c_mod=*/(short)0, acc, /*reuse_a=*/false, /*reuse_b=*/false);

#pragma unroll
            for (int v = 0; v < 8; ++v)
                rmin[v] = fminf(rmin[v], acc[v]);   // min-only inner epilogue
        }
        __syncthreads();
    }

    // deferred ||x||^2: min_m(acc + x2) == min_m(acc) + x2
#pragma unroll
    for (int v = 0; v < 8; ++v) rmin[v] += x2s[wave][hi * 8 + v];

    // min-reduce each row across the 16 lanes of its half (xor < 16 stays in half)
#pragma unroll
    for (int v = 0; v < 8; ++v) {
        float r = rmin[v];
#pragma unroll
        for (int off = 1; off < 16; off <<= 1)
            r = fminf(r, __shfl_xor(r, off, 32));
        rmin[v] = r;
    }
    float s = 0.0f;
#pragma unroll
    for (int v = 0; v < 8; ++v) s += rmin[v];
    s += __shfl_xor(s, 16, 32);  // rows 0-7 (half 0) + rows 8-15 (half 1)

    if (lane == 0) {
        const float scale = (float)(*param) * invDenom;
        atomicAdd(out + slot, s * scale);
    }
}

__global__ void zero_out_kernel(float* out)
{
    if (threadIdx.x < 2) out[threadIdx.x] = 0.0f;
}

extern "C" void kernel_launch(void* const* d_in, const int* in_sizes, int n_in,
                              void* d_out, int out_size, void* d_ws, size_t ws_size,
                              hipStream_t stream)
{
    const float* coarse = (const float*)d_in[0];  // [B,1024,3]
    const float* fine   = (const float*)d_in[1];  // [B,8192,3]
    const float* gt     = (const float*)d_in[2];  // [B,8192,3]
    const int*   pC     = (const int*)d_in[3];    // param_coarse (scalar)
    const int*   pF     = (const int*)d_in[4];    // param_fine (scalar)
    float* out = (float*)d_out;                   // [2]

    const int Nc = 1024, Nf = 8192, Ng = 8192;
    const int B = in_sizes[0] / (Nc * 3);

    zero_out_kernel<<<1, 32, 0, stream>>>(out);

    const dim3 blk(128);
    // loss_coarse = (mean_n min_m + mean_m min_n) of d(coarse,gt), * pC
    chamfer_rowmin_kernel<<<B * Nc / (16 * WAVES), blk, 0, stream>>>(
        coarse, gt, Nc, Ng, pC, 1.0f / (float)(B * Nc), out, 0);
    chamfer_rowmin_kernel<<<B * Ng / (16 * WAVES), blk, 0, stream>>>(
        gt, coarse, Ng, Nc, pC, 1.0f / (float)(B * Ng), out, 0);
    // loss_fine
    chamfer_rowmin_kernel<<<B * Nf / (16 * WAVES), blk, 0, stream>>>(
        fine, gt, Nf, Ng, pF, 1.0f / (float)(B * Nf), out, 1);
    chamfer_rowmin_kernel<<<B * Ng / (16 * WAVES), blk, 0, stream>>>(
        gt, fine, Ng, Nf, pF, 1.0f / (float)(B * Ng), out, 1);
}